// BioBERT_ARG_GNN_61976378081690
// MI455X (gfx1250) — compile-verified
//
#include <hip/hip_runtime.h>

typedef float v2f __attribute__((ext_vector_type(2)));
typedef float v8f __attribute__((ext_vector_type(8)));

#define B_  64
#define S_  512
#define H_  768
#define N_  128
#define E_  1024
#define GH_ 128
#define FC_ 256
#define NL_ 2
#define HC  192   // H chunk for pooling kernel (H_/4)
#define KC  128   // K chunk for LDS-staged GEMM

// LDS byte offsets for gcn_kernel (dynamic shared, no static LDS in kernel)
#define OFF_BUFA 0u
#define OFF_BUFB 65536u
#define OFF_WS   131072u

// ---------------------------------------------------------------------------
// Kernel 1: rat_probs[b,s] = sigmoid(dot(last_hidden[b,s,:], w_rat) + b_rat)
// One wave (32 lanes) per (b,s) row; 8 rows per 256-thread block.
// ---------------------------------------------------------------------------
__global__ __launch_bounds__(256)
void rat_kernel(const float* __restrict__ lh, const float* __restrict__ w_rat,
                const float* __restrict__ b_rat, float* __restrict__ rat) {
    int wave = threadIdx.x >> 5;
    int lane = threadIdx.x & 31;
    int row  = blockIdx.x * 8 + wave;            // [0, B_*S_)
    const float* p = lh + (size_t)row * H_;
    float s = 0.f;
    #pragma unroll 4
    for (int h = lane; h < H_; h += 32) s += p[h] * w_rat[h];
    #pragma unroll
    for (int off = 16; off; off >>= 1) s += __shfl_xor(s, off, 32);
    if (lane == 0) rat[row] = 1.f / (1.f + __expf(-(s + b_rat[0])));
}

// ---------------------------------------------------------------------------
// Kernel 2: segment-mean pooling of seq = last_hidden * rat into x[b, n, h].
// Grid (B_, H_/HC); block = HC threads, each owning one h-column of the chunk.
// Sequential s-loop -> no races on LDS accumulators.
// ---------------------------------------------------------------------------
__global__ __launch_bounds__(HC)
void pool_kernel(const float* __restrict__ lh, const int* __restrict__ sub,
                 const float* __restrict__ rat, float* __restrict__ xg) {
    extern __shared__ float smem[];
    float* sums = smem;                    // N_*HC floats (96 KB)
    int*   subS = (int*)(smem + N_ * HC);  // S_ ints
    int*   cntS = subS + S_;               // N_ ints

    int b  = blockIdx.x;
    int hb = blockIdx.y * HC;
    int t  = threadIdx.x;

    for (int i = t; i < N_ * HC; i += HC) sums[i] = 0.f;
    for (int i = t; i < S_; i += HC)      subS[i] = sub[b * S_ + i];
    __syncthreads();

    const float* lhb  = lh + (size_t)b * S_ * H_ + hb;
    const float* ratb = rat + b * S_;
    for (int s = 0; s < S_; ++s) {
        int   w = subS[s];
        float p = ratb[s];
        sums[w * HC + t] += p * lhb[(size_t)s * H_ + t];
    }
    __syncthreads();
    if (t < N_) {
        int c = 0;
        for (int s = 0; s < S_; ++s) c += (subS[s] == t);
        cntS[t] = c;
    }
    __syncthreads();
    for (int i = t; i < N_ * HC; i += HC) {
        int n = i / HC, h = i - n * HC;
        int c = cntS[n];
        float sc = (c > 0) ? (1.f / (float)c) : 0.f;
        xg[(size_t)b * N_ * H_ + (size_t)n * H_ + hb + h] = sums[i] * sc;
    }
}

// ---------------------------------------------------------------------------
// Async stage helper: copy 16 bytes per call, lane-granular, global -> LDS.
// Tracked with ASYNCcnt; caller must s_wait_asynccnt + __syncthreads before use.
// ---------------------------------------------------------------------------
__device__ __forceinline__ void async_copy16(unsigned lds_byte_off, const float* g) {
    asm volatile("global_load_async_to_lds_b128 %0, %1, off"
                 :: "v"(lds_byte_off), "v"(g) : "memory");
}
__device__ __forceinline__ void async_wait0() {
    asm volatile("s_wait_asynccnt 0x0" ::: "memory");
}

// ---------------------------------------------------------------------------
// Kernel 3: per-graph 2-layer GCN + mean readout. One 256-thread WG per graph.
// GEMMs via V_WMMA_F32_16X16X4_F32 with K-chunked LDS staging done by
// GLOBAL_LOAD_ASYNC_TO_LDS_B128. Message passing with LDS float atomics.
// ---------------------------------------------------------------------------
__global__ __launch_bounds__(256)
void gcn_kernel(const float* __restrict__ xg, const int* __restrict__ eidx,
                const float* __restrict__ Wg1, const float* __restrict__ bg1,
                const float* __restrict__ Wg2, const float* __restrict__ bg2,
                float* __restrict__ gfeat) {
    extern __shared__ float smem[];
    float* bufA = smem;                    // 16384 floats: GEMM out / MP source
    float* bufB = smem + 16384;            // 16384 floats: x-stage / x1 / MP dest
    float* Ws   = smem + 32768;            // 16384 floats: W-chunk stage
    float* dinv = smem + 49152;            // N_ floats
    int*   esrc = (int*)(dinv + N_);       // E_ ints
    int*   edst = esrc + E_;               // E_ ints
    int*   degI = edst + E_;               // N_ ints

    int b    = blockIdx.x;
    int tid  = threadIdx.x;
    int wave = tid >> 5;
    int lane = tid & 31;
    int half = lane >> 4;
    int r    = lane & 15;
    int mt   = wave;                       // 8 waves -> 8 M-tiles of 16

    // ---- edges + degrees (self-loops included: deg starts at 1) ----
    for (int e = tid; e < E_; e += 256) {
        esrc[e] = eidx[(size_t)b * 2 * E_ + e];
        edst[e] = eidx[(size_t)b * 2 * E_ + E_ + e];
    }
    if (tid < N_) degI[tid] = 1;
    __syncthreads();
    for (int e = tid; e < E_; e += 256) atomicAdd(&degI[edst[e]], 1);
    __syncthreads();
    if (tid < N_) dinv[tid] = rsqrtf((float)degI[tid]);

    // ---- GEMM1: bufA(128x128) = x_b(128x768) @ Wg1(768x128) ----
    // K chunked by KC=128; operands staged to LDS with async-to-LDS copies.
    const float* xb = xg + (size_t)b * N_ * H_;
    v8f acc[8];
    #pragma unroll
    for (int nt = 0; nt < 8; ++nt) acc[nt] = (v8f){};

    for (int kc = 0; kc < H_ / KC; ++kc) {
        // stage x[:, kc*KC .. +KC) into bufB (row-major 128xKC), 16B per lane
        for (int i = tid * 4; i < N_ * KC; i += 256 * 4) {
            int row = i >> 7, col = i & (KC - 1);
            async_copy16(OFF_BUFB + (unsigned)i * 4u,
                         xb + (size_t)row * H_ + kc * KC + col);
        }
        // stage Wg1 rows [kc*KC, +KC) (contiguous 64KB) into Ws
        for (int i = tid * 4; i < KC * GH_; i += 256 * 4) {
            async_copy16(OFF_WS + (unsigned)i * 4u,
                         Wg1 + (size_t)kc * KC * GH_ + i);
        }
        async_wait0();
        __syncthreads();

        for (int nt = 0; nt < 8; ++nt) {
            v8f c = acc[nt];
            for (int kk = 0; kk < KC / 4; ++kk) {
                int k0 = kk * 4 + half * 2;
                v2f a, w;
                a.x = bufB[(mt * 16 + r) * KC + k0];
                a.y = bufB[(mt * 16 + r) * KC + k0 + 1];
                w.x = Ws[k0 * GH_ + nt * 16 + r];
                w.y = Ws[(k0 + 1) * GH_ + nt * 16 + r];
                c = __builtin_amdgcn_wmma_f32_16x16x4_f32(
                    false, a, false, w, (short)0, c, false, false);
            }
            acc[nt] = c;
        }
        __syncthreads();   // protect bufB/Ws before next chunk restage
    }
    #pragma unroll
    for (int nt = 0; nt < 8; ++nt)
        #pragma unroll
        for (int j = 0; j < 8; ++j)
            bufA[(mt * 16 + j + half * 8) * GH_ + nt * 16 + r] = acc[nt][j];
    __syncthreads();

    // ---- MP1: bufB = relu(bias + self + edge messages) ----
    for (int i = tid; i < N_ * GH_; i += 256) {
        int n = i >> 7;
        bufB[i] = bg1[i & 127] + bufA[i] * dinv[n] * dinv[n];
    }
    __syncthreads();
    for (int i = tid; i < E_ * GH_; i += 256) {
        int e = i >> 7, c = i & 127;
        int s = esrc[e], d = edst[e];
        atomicAdd(&bufB[d * GH_ + c], bufA[s * GH_ + c] * dinv[s] * dinv[d]);
    }
    __syncthreads();
    for (int i = tid; i < N_ * GH_; i += 256) bufB[i] = fmaxf(bufB[i], 0.f);

    // ---- GEMM2: bufA = bufB(128x128, LDS) @ Wg2(128x128) ----
    // Stage the entire Wg2 (64 KB) into Ws asynchronously.
    for (int i = tid * 4; i < GH_ * GH_; i += 256 * 4)
        async_copy16(OFF_WS + (unsigned)i * 4u, Wg2 + i);
    async_wait0();
    __syncthreads();   // also orders the relu writes above

    for (int nt = 0; nt < 8; ++nt) {
        v8f c = {};
        for (int kk = 0; kk < GH_ / 4; ++kk) {
            int k0 = kk * 4 + half * 2;
            v2f a, w;
            a.x = bufB[(mt * 16 + r) * GH_ + k0];
            a.y = bufB[(mt * 16 + r) * GH_ + k0 + 1];
            w.x = Ws[k0 * GH_ + nt * 16 + r];
            w.y = Ws[(k0 + 1) * GH_ + nt * 16 + r];
            c = __builtin_amdgcn_wmma_f32_16x16x4_f32(
                false, a, false, w, (short)0, c, false, false);
        }
        #pragma unroll
        for (int j = 0; j < 8; ++j)
            bufA[(mt * 16 + j + half * 8) * GH_ + nt * 16 + r] = c[j];
    }
    __syncthreads();

    // ---- MP2 into bufB, then mean over nodes (relu folded in) ----
    for (int i = tid; i < N_ * GH_; i += 256) {
        int n = i >> 7;
        bufB[i] = bg2[i & 127] + bufA[i] * dinv[n] * dinv[n];
    }
    __syncthreads();
    for (int i = tid; i < E_ * GH_; i += 256) {
        int e = i >> 7, c = i & 127;
        int s = esrc[e], d = edst[e];
        atomicAdd(&bufB[d * GH_ + c], bufA[s * GH_ + c] * dinv[s] * dinv[d]);
    }
    __syncthreads();
    if (tid < GH_) {
        float sum = 0.f;
        for (int n = 0; n < N_; ++n) sum += fmaxf(bufB[n * GH_ + tid], 0.f);
        gfeat[b * GH_ + tid] = sum * (1.f / (float)N_);
    }
}

// ---------------------------------------------------------------------------
// Kernel 4: FC head. h = relu([cls, gfeat] @ Wfc1 + b); logits = h @ Wfc2 + b.
// One block per batch row, 256 threads (one per FC_ column).
// ---------------------------------------------------------------------------
__global__ __launch_bounds__(FC_)
void head_kernel(const float* __restrict__ lh, const float* __restrict__ gfeat,
                 const float* __restrict__ Wfc1, const float* __restrict__ bfc1,
                 const float* __restrict__ Wfc2, const float* __restrict__ bfc2,
                 float* __restrict__ logits) {
    __shared__ float hbuf[FC_];
    int b = blockIdx.x, t = threadIdx.x;
    const float* cls = lh + (size_t)b * S_ * H_;   // last_hidden[b, 0, :]
    const float* gf  = gfeat + b * GH_;
    float s = bfc1[t];
    for (int k = 0; k < H_; ++k)  s += cls[k] * Wfc1[(size_t)k * FC_ + t];
    for (int k = 0; k < GH_; ++k) s += gf[k] * Wfc1[(size_t)(H_ + k) * FC_ + t];
    hbuf[t] = fmaxf(s, 0.f);
    __syncthreads();
    if (t < NL_) {
        float a = bfc2[t];
        for (int k = 0; k < FC_; ++k) a += hbuf[k] * Wfc2[k * NL_ + t];
        logits[b * NL_ + t] = a;
    }
}

// ---------------------------------------------------------------------------
extern "C" void kernel_launch(void* const* d_in, const int* in_sizes, int n_in,
                              void* d_out, int out_size, void* d_ws, size_t ws_size,
                              hipStream_t stream) {
    const float* lh   = (const float*)d_in[0];
    const int*   sub  = (const int*)  d_in[1];
    const int*   eidx = (const int*)  d_in[2];
    const float* wrat = (const float*)d_in[3];
    const float* brat = (const float*)d_in[4];
    const float* Wg1  = (const float*)d_in[5];
    const float* bg1  = (const float*)d_in[6];
    const float* Wg2  = (const float*)d_in[7];
    const float* bg2  = (const float*)d_in[8];
    const float* Wfc1 = (const float*)d_in[9];
    const float* bfc1 = (const float*)d_in[10];
    const float* Wfc2 = (const float*)d_in[11];
    const float* bfc2 = (const float*)d_in[12];

    float* out    = (float*)d_out;
    float* logits = out;               // B_*NL_ = 128 floats
    float* rat    = out + B_ * NL_;    // B_*S_  = 32768 floats

    // Workspace: x[b,n,h] (24 MB) + gfeat[b,c]
    float* xg    = (float*)d_ws;
    float* gfeat = xg + (size_t)B_ * N_ * H_;

    rat_kernel<<<B_ * S_ / 8, 256, 0, stream>>>(lh, wrat, brat, rat);

    size_t smem2 = (size_t)(N_ * HC) * 4 + S_ * 4 + N_ * 4;      // ~98.5 KB
    pool_kernel<<<dim3(B_, H_ / HC), HC, smem2, stream>>>(lh, sub, rat, xg);

    // bufA + bufB + Ws + dinv + esrc + edst + degI
    size_t smem3 = (size_t)(3 * N_ * GH_ + N_) * 4 + (2 * E_ + N_) * 4;  // ~201 KB
    gcn_kernel<<<B_, 256, smem3, stream>>>(xg, eidx, Wg1, bg1, Wg2, bg2, gfeat);

    head_kernel<<<B_, FC_, 0, stream>>>(lh, gfeat, Wfc1, bfc1, Wfc2, bfc2, logits);
}